// GraphAttentionLayer_16174846837124
// MI455X (gfx1250) — compile-verified
//
#include <hip/hip_runtime.h>
#include <math.h>

#define LRELU_ALPHA 0.2f
#define IN_F   128
#define OUT_F  64
#define HEADS  4
#define NHO    (HEADS * OUT_F)   // 256 features per node in xt / out

typedef __attribute__((ext_vector_type(2))) float v2f;
typedef __attribute__((ext_vector_type(8))) float v8f;

__device__ __forceinline__ float leaky(float v) {
    return v > 0.0f ? v : LRELU_ALPHA * v;
}

// float atomic max via monotone int/uint bit mapping (works incl. -inf init)
__device__ __forceinline__ void atomicMaxF32(float* addr, float val) {
    if (val >= 0.0f) atomicMax((int*)addr, __float_as_int(val));
    else             atomicMin((unsigned int*)addr, __float_as_uint(val));
}

// ---------------- init: out = 0, max_e = -inf, sum_e = 0 ----------------
__global__ void gat_init(float* __restrict__ out, float* __restrict__ max_e,
                         float* __restrict__ sum_e, int n_out, int n_nh) {
    int i = blockIdx.x * blockDim.x + threadIdx.x;
    if (i < n_out) out[i] = 0.0f;
    if (i < n_nh) { max_e[i] = -__builtin_huge_valf(); sum_e[i] = 0.0f; }
}

// ---------------- xt[n,h,o] = sum_i x[n,i] * W[h,i,o] via WMMA f32 ------
// block = 512 threads = 16 waves; wave w -> (head = w/4, o-tile = w%4)
// each wave computes a 16(node) x 16(out) f32 tile with K=128 in 32 steps
__global__ __launch_bounds__(512) void gat_gemm_wmma(
    const float* __restrict__ x, const float* __restrict__ W,
    float* __restrict__ xt, int n_nodes) {
    const int lane  = threadIdx.x & 31;
    const int wave  = threadIdx.x >> 5;        // 0..15
    const int h     = wave >> 2;
    const int o0    = (wave & 3) << 4;
    const int n0    = blockIdx.x << 4;
    const int mrow  = lane & 15;               // M / N position within tile
    const int khalf = lane >> 4;               // 0 or 1 (K sub-slot)

    int nload = n0 + mrow; if (nload > n_nodes - 1) nload = n_nodes - 1;
    const float* __restrict__ xrow = x + (size_t)nload * IN_F + 2 * khalf;
    const float* __restrict__ Wh   = W + (size_t)h * IN_F * OUT_F + o0 + mrow;

    v8f c = {};
    #pragma unroll 4
    for (int k = 0; k < IN_F; k += 4) {
        // A 16x4 f32: lane holds A[M=mrow, K = k + 2*khalf + {0,1}]
        v2f a;
        a.x = xrow[k];
        a.y = xrow[k + 1];
        // B 4x16 f32: lane holds B[K = k + 2*khalf + {0,1}, N=mrow]
        v2f b;
        b.x = Wh[(size_t)(k + 2 * khalf + 0) * OUT_F];
        b.y = Wh[(size_t)(k + 2 * khalf + 1) * OUT_F];
        c = __builtin_amdgcn_wmma_f32_16x16x4_f32(false, a, false, b,
                                                  (short)0, c, false, false);
    }

    // C 16x16 f32: VGPR v holds C[M = v + 8*khalf, N = mrow]
    float* __restrict__ op = xt + (size_t)n0 * NHO + h * OUT_F + o0 + mrow +
                             (size_t)(8 * khalf) * NHO;
    if (n0 + 16 <= n_nodes) {
        // full tile: straight-line stores, no per-row EXEC juggling
        #pragma unroll
        for (int v = 0; v < 8; ++v)
            op[(size_t)v * NHO] = c[v];
    } else {
        // tail tile only
        #pragma unroll
        for (int v = 0; v < 8; ++v)
            if (n0 + v + 8 * khalf < n_nodes)
                op[(size_t)v * NHO] = c[v];
    }
}

// -------- per-node attention scalars: alpha = xt[n,h,:] . a[h,:] --------
__global__ void gat_alpha(const float* __restrict__ xt,
                          const float* __restrict__ a,
                          float* __restrict__ alpha_src,
                          float* __restrict__ alpha_dst, int n_nh) {
    int idx = blockIdx.x * blockDim.x + threadIdx.x;  // node*HEADS + h
    if (idx >= n_nh) return;
    int h = idx & (HEADS - 1);
    const float* __restrict__ row = xt + (size_t)(idx >> 2) * NHO + h * OUT_F;
    const float* __restrict__ as  = a + (size_t)h * 2 * OUT_F;   // a[h,0:64]
    const float* __restrict__ ad  = as + OUT_F;                  // a[h,64:128]
    float s = 0.0f, d = 0.0f;
    #pragma unroll 8
    for (int o = 0; o < OUT_F; ++o) {
        float v = row[o];
        s += v * as[o];
        d += v * ad[o];
    }
    alpha_src[idx] = s;
    alpha_dst[idx] = d;
}

// -------- segment max over dst --------
__global__ void gat_edge_max(const long long* __restrict__ ei,
                             const float* __restrict__ alpha_src,
                             const float* __restrict__ alpha_dst,
                             float* __restrict__ max_e,
                             int n_edges, int n_eh) {
    int idx = blockIdx.x * blockDim.x + threadIdx.x;  // edge*HEADS + h
    if (idx >= n_eh) return;
    int e = idx >> 2, h = idx & (HEADS - 1);
    long long s = ei[e], d = ei[n_edges + e];
    float ev = leaky(alpha_src[s * HEADS + h] + alpha_dst[d * HEADS + h]);
    atomicMaxF32(max_e + d * HEADS + h, ev);
}

// -------- segment sum of exp(e - max) over dst --------
__global__ void gat_edge_sum(const long long* __restrict__ ei,
                             const float* __restrict__ alpha_src,
                             const float* __restrict__ alpha_dst,
                             const float* __restrict__ max_e,
                             float* __restrict__ sum_e,
                             int n_edges, int n_eh) {
    int idx = blockIdx.x * blockDim.x + threadIdx.x;
    if (idx >= n_eh) return;
    int e = idx >> 2, h = idx & (HEADS - 1);
    long long s = ei[e], d = ei[n_edges + e];
    float ev = leaky(alpha_src[s * HEADS + h] + alpha_dst[d * HEADS + h]);
    float x  = expf(ev - max_e[d * HEADS + h]);
    atomicAdd(sum_e + d * HEADS + h, x);
}

// -------- weighted scatter: out[dst] += attention * xt[src] --------
// one block (256 threads) per edge; thread t covers feature t = h*64 + o
__global__ __launch_bounds__(NHO) void gat_scatter(
    const long long* __restrict__ ei, const float* __restrict__ xt,
    const float* __restrict__ alpha_src, const float* __restrict__ alpha_dst,
    const float* __restrict__ max_e, const float* __restrict__ sum_e,
    float* __restrict__ out, int n_edges) {
    int e = blockIdx.x;
    int t = threadIdx.x;               // 0..255
    int h = t >> 6;
    long long s = ei[e], d = ei[n_edges + e];
    float ev  = leaky(alpha_src[s * HEADS + h] + alpha_dst[d * HEADS + h]);
    float att = expf(ev - max_e[d * HEADS + h]) /
                (sum_e[d * HEADS + h] + 1e-10f);
    atomicAdd(out + (size_t)d * NHO + t, xt[(size_t)s * NHO + t] * att);
}

extern "C" void kernel_launch(void* const* d_in, const int* in_sizes, int n_in,
                              void* d_out, int out_size, void* d_ws, size_t ws_size,
                              hipStream_t stream) {
    const float*     x  = (const float*)d_in[0];
    const long long* ei = (const long long*)d_in[1];   // int64 edge_index [2, E]
    const float*     W  = (const float*)d_in[2];       // [4,128,64]
    const float*     a  = (const float*)d_in[3];       // [4,128,1]
    float* out = (float*)d_out;                        // [n_nodes, 256]

    const int n_nodes = in_sizes[0] / IN_F;            // 50000
    const int n_edges = in_sizes[1] / 2;               // 800000

    // workspace layout
    float* xt        = (float*)d_ws;                           // n*256
    float* alpha_src = xt + (size_t)n_nodes * NHO;             // n*4
    float* alpha_dst = alpha_src + (size_t)n_nodes * HEADS;    // n*4
    float* max_e     = alpha_dst + (size_t)n_nodes * HEADS;    // n*4
    float* sum_e     = max_e + (size_t)n_nodes * HEADS;        // n*4

    const int n_out = n_nodes * NHO;
    const int n_nh  = n_nodes * HEADS;
    const int n_eh  = n_edges * HEADS;

    gat_init<<<(n_out + 255) / 256, 256, 0, stream>>>(out, max_e, sum_e,
                                                      n_out, n_nh);
    gat_gemm_wmma<<<(n_nodes + 15) / 16, 512, 0, stream>>>(x, W, xt, n_nodes);
    gat_alpha<<<(n_nh + 255) / 256, 256, 0, stream>>>(xt, a, alpha_src,
                                                      alpha_dst, n_nh);
    gat_edge_max<<<(n_eh + 255) / 256, 256, 0, stream>>>(ei, alpha_src,
                                                         alpha_dst, max_e,
                                                         n_edges, n_eh);
    gat_edge_sum<<<(n_eh + 255) / 256, 256, 0, stream>>>(ei, alpha_src,
                                                         alpha_dst, max_e,
                                                         sum_e, n_edges, n_eh);
    gat_scatter<<<n_edges, NHO, 0, stream>>>(ei, xt, alpha_src, alpha_dst,
                                             max_e, sum_e, out, n_edges);
}